// SparseAdam_54305566490904
// MI455X (gfx1250) — compile-verified
//
#include <hip/hip_runtime.h>
#include <stdint.h>

// Sparse Adam embedding update for MI455X (gfx1250, wave32).
// - One wave == one 128-float embedding row (32 lanes x float4).
// - Duplicate-grad accumulation via f32 global atomics into d_out (L2-resident).
// - Embedding table streamed HBM->LDS with the Tensor Data Mover
//   (tensor_load_to_lds), double-buffered per workgroup, synced with
//   s_wait_tensorcnt + workgroup barriers.
// - Non-temporal loads/stores on all streamed data.

#define D_DIM     128
#define TILE_ROWS 16   // rows per TDM tile == waves per block

typedef float    v4f __attribute__((ext_vector_type(4)));
typedef unsigned v4u __attribute__((ext_vector_type(4)));
typedef int      v4i __attribute__((ext_vector_type(4)));
typedef int      v8i __attribute__((ext_vector_type(8)));

__device__ __forceinline__ v4f nt_load4(const float* p) {
  return __builtin_nontemporal_load((const v4f*)p);
}
__device__ __forceinline__ void nt_store4(float* p, v4f v) {
  __builtin_nontemporal_store(v, (v4f*)p);
}

// Issue one TDM 2D tile load: TILE_ROWS x D_DIM f32 from a row-major
// [n_rows x D_DIM] tensor (tile starts at gtile) into LDS at lds_byte.
// D# layout per CDNA5 ISA ch.8 (group0: count/lds/global/type,
// group1: data_size, tensor dims, tile dims, dim0 stride).
__device__ __forceinline__ void tdm_load_tile(const float* gtile,
                                              unsigned lds_byte,
                                              unsigned n_rows)
{
  unsigned long long ga = (unsigned long long)(uintptr_t)gtile;
  v4u g0;
  g0[0] = 1u;                                               // count=1, user mode
  g0[1] = lds_byte;                                         // lds_addr (bytes)
  g0[2] = (unsigned)ga;                                     // global_addr[31:0]
  g0[3] = (unsigned)((ga >> 32) & 0x01FFFFFFull)            // global_addr[56:32]
        | (2u << 30);                                       // type=2 ("image")
  const unsigned td0 = (unsigned)D_DIM;                     // tensor_dim0 (elems)
  const unsigned td1 = n_rows;                              // tensor_dim1 (rows)
  v8i g1;
  g1[0] = (int)(2u << 16);                                  // data_size=2 -> 4B
  g1[1] = (int)((td0 & 0xFFFFu) << 16);                     // tensor_dim0[15:0]
  g1[2] = (int)((td0 >> 16) | ((td1 & 0xFFFFu) << 16));     // dim0 hi | dim1 lo
  g1[3] = (int)((td1 >> 16) | ((unsigned)D_DIM << 16));     // dim1 hi | tile_dim0
  g1[4] = (int)TILE_ROWS;                                   // tile_dim1; tile_dim2=0
  g1[5] = (int)D_DIM;                                       // tensor_dim0_stride lo
  g1[6] = 0;                                                // stride0 hi | stride1 lo
  g1[7] = 0;                                                // stride1 hi
  v4i z4 = {0, 0, 0, 0};                                    // groups 2/3 unused (2D)
#if defined(__clang_major__) && (__clang_major__ >= 23)
  v8i z8 = {0, 0, 0, 0, 0, 0, 0, 0};
  __builtin_amdgcn_tensor_load_to_lds(g0, g1, z4, z4, z8, 0);
#else
  __builtin_amdgcn_tensor_load_to_lds(g0, g1, z4, z4, 0);
#endif
}

// Kernel 1: zero the per-row duplicate counts in workspace.
__global__ void zero_counts_kernel(unsigned* __restrict__ counts, int n_emb)
{
  int i = blockIdx.x * blockDim.x + threadIdx.x;
  if (i < n_emb) counts[i] = 0u;
}

// Kernel 2: one wave per occurrence -- zero the touched row of d_out
// (gsum accumulator; duplicate zero-writes are benign) and count duplicates.
__global__ void zero_touched_kernel(const int* __restrict__ idx,
                                    float* __restrict__ out,
                                    unsigned* __restrict__ counts,
                                    int n_idx)
{
  const int tid  = blockIdx.x * blockDim.x + threadIdx.x;
  const int occ  = tid >> 5;
  const int lane = tid & 31;
  if (occ >= n_idx) return;
  const int row = idx[occ];
  v4f z = {0.0f, 0.0f, 0.0f, 0.0f};
  nt_store4(out + (size_t)row * D_DIM + (size_t)lane * 4, z);
  if (lane == 0) atomicAdd(counts + row, 1u);
}

// Kernel 3: one wave per occurrence -- accumulate grads into d_out with
// f32 global atomics (no-return -> STOREcnt path, resolved at L2).
__global__ void scatter_grads_kernel(const int* __restrict__ idx,
                                     const float* __restrict__ grad,
                                     float* __restrict__ out,
                                     int n_idx)
{
  const int tid  = blockIdx.x * blockDim.x + threadIdx.x;
  const int occ  = tid >> 5;
  const int lane = tid & 31;
  if (occ >= n_idx) return;
  const int row = idx[occ];
  v4f g = nt_load4(grad + (size_t)occ * D_DIM + (size_t)lane * 4);
  float* dst = out + (size_t)row * D_DIM + (size_t)lane * 4;
  atomicAdd(dst + 0, g[0]);
  atomicAdd(dst + 1, g[1]);
  atomicAdd(dst + 2, g[2]);
  atomicAdd(dst + 3, g[3]);
}

// Kernel 4: fused sparse Adam + full-table writeback.
// emb tiles streamed via TDM into double-buffered LDS; wave w handles row
// tile*16+w, lane handles 4 columns.
__global__ void __launch_bounds__(TILE_ROWS * 32)
sparse_adam_finalize(const float* __restrict__ emb,
                     const float* __restrict__ step_cnt,
                     const float* __restrict__ mem,
                     const float* __restrict__ pwr,
                     const unsigned* __restrict__ counts,
                     float* __restrict__ out,
                     int n_emb, int ntiles)
{
  __shared__ float lds[2][TILE_ROWS][D_DIM];   // 2 x 8 KB
  const int wave = threadIdx.x >> 5;
  const int lane = threadIdx.x & 31;

  int t = blockIdx.x;
  if (t >= ntiles) return;

  // Prologue: wave 0 kicks off the first tile's TDM transfer.
  if (wave == 0)
    tdm_load_tile(emb + (size_t)t * TILE_ROWS * D_DIM,
                  (unsigned)(uintptr_t)&lds[0][0][0], (unsigned)n_emb);

  int buf = 0;
  for (; t < ntiles; t += (int)gridDim.x) {
    if (wave == 0) __builtin_amdgcn_s_wait_tensorcnt(0);   // tile t landed in LDS
    __syncthreads();                                       // publish to all waves

    const int tn = t + (int)gridDim.x;
    if (wave == 0 && tn < ntiles)                          // prefetch next tile
      tdm_load_tile(emb + (size_t)tn * TILE_ROWS * D_DIM,
                    (unsigned)(uintptr_t)&lds[buf ^ 1][0][0], (unsigned)n_emb);

    const int r = t * TILE_ROWS + wave;
    if (r < n_emb) {
      v4f e = *(const v4f*)&lds[buf][wave][lane * 4];      // ds_load_b128
      const unsigned c = counts[r];
      v4f res = e;
      if (c != 0u) {
        const size_t off   = (size_t)r * D_DIM + (size_t)lane * 4;
        const float  inv_c = 1.0f / (float)c;
        v4f gs = *(const v4f*)(out + off);                 // gsum (hot in L2)
        v4f m  = nt_load4(mem + off);
        v4f p  = nt_load4(pwr + off);
        const float stp = step_cnt[r] + 1.0f;
        const float d1  = 1.0f - __powf(0.9f,   stp);
        const float d2  = 1.0f - __powf(0.999f, stp);
        #pragma unroll
        for (int k = 0; k < 4; ++k) {
          float g   = gs[k] * inv_c;                       // mean of duplicates
          float mn  = 0.9f   * m[k] + 0.1f   * g;
          float vn  = 0.999f * p[k] + 0.001f * g * g;
          float upd = 1e-3f * (mn / d1) / (sqrtf(vn / d2) + 1e-8f);
          res[k] = e[k] - upd;
        }
      }
      nt_store4(out + (size_t)r * D_DIM + (size_t)lane * 4, res);
    }
    buf ^= 1;
    // Next iteration's leading barrier orders buffer reuse; no trailing
    // barrier needed (TDM into this buf is issued only after it).
  }
}

extern "C" void kernel_launch(void* const* d_in, const int* in_sizes, int n_in,
                              void* d_out, int out_size, void* d_ws, size_t ws_size,
                              hipStream_t stream)
{
  const int*   idx  = (const int*)  d_in[0];
  const float* grad = (const float*)d_in[1];
  const float* emb  = (const float*)d_in[2];
  const float* stp  = (const float*)d_in[3];
  const float* mem  = (const float*)d_in[4];
  const float* pwr  = (const float*)d_in[5];
  float*       out  = (float*)d_out;
  unsigned*    counts = (unsigned*)d_ws;     // n_emb * 4 bytes (~1 MB)

  const int n_idx = in_sizes[0];             // 65536
  const int n_emb = in_sizes[3];             // 250000 (state_step length)
  const int ntiles = (n_emb + TILE_ROWS - 1) / TILE_ROWS;

  zero_counts_kernel<<<(n_emb + 255) / 256, 256, 0, stream>>>(counts, n_emb);

  const int occ_threads = n_idx * 32;
  zero_touched_kernel<<<(occ_threads + 255) / 256, 256, 0, stream>>>(
      idx, out, counts, n_idx);

  scatter_grads_kernel<<<(occ_threads + 255) / 256, 256, 0, stream>>>(
      idx, grad, out, n_idx);

  sparse_adam_finalize<<<1024, TILE_ROWS * 32, 0, stream>>>(
      emb, stp, mem, pwr, counts, out, n_emb, ntiles);
}